// L2_ChamferLoss_19164144075462
// MI455X (gfx1250) — compile-verified
//
#include <hip/hip_runtime.h>

typedef __attribute__((ext_vector_type(2))) float v2f;
typedef __attribute__((ext_vector_type(8))) float v8f;

#define TILE    16
#define CHUNK   2048      // reference points staged in LDS per round (32 KB)
#define THREADS 256       // 8 waves; each wave owns a 16-row tile

// One pass of chamfer: for each point in P, min squared distance to any point in Q.
// P: [B,N,3], Q: [B,M,3], outMin: [B,N].   N multiple of 128, M multiple of CHUNK.
__global__ __launch_bounds__(THREADS) void chamfer_min_wmma(
    const float* __restrict__ P,
    const float* __restrict__ Q,
    float* __restrict__ outMin,
    int N, int M)
{
    __shared__ float4 qs[CHUNK];   // (x, y, z, x^2+y^2+z^2)

    const int rowBlocks = N / 128;
    const int b    = blockIdx.x / rowBlocks;
    const int rb   = blockIdx.x % rowBlocks;
    const int tid  = threadIdx.x;
    const int wave = tid >> 5;
    const int lane = tid & 31;
    const int half = lane >> 4;    // 0: lanes 0-15 (K=0,1) | 1: lanes 16-31 (K=2,3)
    const int l15  = lane & 15;
    const int rowbase = rb * 128 + wave * TILE;

    const float* Pb = P + (size_t)b * N * 3;
    const float* Qb = Q + (size_t)b * M * 3;

    // ---- A-matrix 16x4 (32-bit layout): lane holds 2 K-values of its row.
    // Row vector = (-2x, -2y, -2z, 1)
    const int myrow = rowbase + l15;
    const float px = Pb[myrow * 3 + 0];
    const float py = Pb[myrow * 3 + 1];
    const float pz = Pb[myrow * 3 + 2];
    const float sq1 = px * px + py * py + pz * pz;
    v2f a;
    if (half == 0) { a.x = -2.0f * px; a.y = -2.0f * py; }
    else           { a.x = -2.0f * pz; a.y = 1.0f;       }

    // C/D layout: VGPR r -> row (rowbase + r + 8*half). Pre-broadcast ||p||^2.
    float cinit[8];
    #pragma unroll
    for (int r = 0; r < 8; ++r)
        cinit[r] = __shfl(sq1, r + half * 8, 32);

    float rowmin[8];
    #pragma unroll
    for (int r = 0; r < 8; ++r) rowmin[r] = 3.4e38f;

    for (int base = 0; base < M; base += CHUNK) {
        // Cooperative LDS fill: (x, y, z, ||q||^2) per reference point.
        for (int i = tid; i < CHUNK; i += THREADS) {
            const float* q = Qb + (size_t)(base + i) * 3;
            const float qx = q[0], qy = q[1], qz = q[2];
            qs[i] = make_float4(qx, qy, qz, qx * qx + qy * qy + qz * qz);
        }
        __syncthreads();

        #pragma unroll 4
        for (int t = 0; t < CHUNK; t += TILE) {
            // B-matrix 4x16: lanes 0-15 load (x,y), lanes 16-31 load (z,||q||^2)
            // of column point (t + l15): one 8-byte LDS load per lane.
            const float* bp =
                reinterpret_cast<const float*>(&qs[t + l15]) + (half << 1);
            const v2f bv = *reinterpret_cast<const v2f*>(bp);

            v8f c;
            #pragma unroll
            for (int r = 0; r < 8; ++r) c[r] = cinit[r];

            // D = ||p||^2 + ||q||^2 - 2 p.q   (full squared distance in the MMA)
            c = __builtin_amdgcn_wmma_f32_16x16x4_f32(
                    false, a, false, bv, (short)0, c, false, false);

            // Running min only; clamp-to-zero is hoisted out of the loop
            // (max(.,0) commutes with min).
            #pragma unroll
            for (int r = 0; r < 8; ++r)
                rowmin[r] = fminf(rowmin[r], c[r]);
        }
        __syncthreads();
    }

    // Min across the 16 lanes of each half (columns of the tile).
    #pragma unroll
    for (int off = 8; off >= 1; off >>= 1) {
        #pragma unroll
        for (int r = 0; r < 8; ++r)
            rowmin[r] = fminf(rowmin[r], __shfl_xor(rowmin[r], off, 32));
    }

    // Lane 0 writes rows rowbase+0..7, lane 16 writes rowbase+8..15.
    if (l15 == 0) {
        #pragma unroll
        for (int r = 0; r < 8; ++r)
            outMin[(size_t)b * N + rowbase + half * 8 + r] =
                fmaxf(rowmin[r], 0.0f);   // deferred clamp of tiny negatives
    }
}

// mean(d1) + mean(d2) -> out[0]. Single block, deterministic tree reduce.
__global__ __launch_bounds__(256) void chamfer_reduce(
    const float* __restrict__ d1, const float* __restrict__ d2,
    float* __restrict__ out, int n1, int n2)
{
    __shared__ float s[256];
    const int tid = threadIdx.x;
    float acc1 = 0.0f, acc2 = 0.0f;
    for (int i = tid; i < n1; i += 256) acc1 += d1[i];
    for (int i = tid; i < n2; i += 256) acc2 += d2[i];
    s[tid] = acc1 / (float)n1 + acc2 / (float)n2;
    __syncthreads();
    for (int o = 128; o > 0; o >>= 1) {
        if (tid < o) s[tid] += s[tid + o];
        __syncthreads();
    }
    if (tid == 0) out[0] = s[0];
}

extern "C" void kernel_launch(void* const* d_in, const int* in_sizes, int n_in,
                              void* d_out, int out_size, void* d_ws, size_t ws_size,
                              hipStream_t stream) {
    const float* a1 = (const float*)d_in[0];   // [B, N, 3]
    const float* a2 = (const float*)d_in[1];   // [B, M, 3]
    const int N = 4096;
    const int M = 4096;
    const int B = in_sizes[0] / (N * 3);

    float* dist1 = (float*)d_ws;               // [B, N]
    float* dist2 = dist1 + (size_t)B * N;      // [B, M]

    // Pass 1: nearest in array2 for each point of array1.
    chamfer_min_wmma<<<dim3(B * (N / 128)), dim3(THREADS), 0, stream>>>(
        a1, a2, dist1, N, M);
    // Pass 2 (symmetric): nearest in array1 for each point of array2.
    chamfer_min_wmma<<<dim3(B * (M / 128)), dim3(THREADS), 0, stream>>>(
        a2, a1, dist2, M, N);

    chamfer_reduce<<<dim3(1), dim3(256), 0, stream>>>(
        dist1, dist2, (float*)d_out, B * N, B * M);
}